// Model_36627481100881
// MI455X (gfx1250) — compile-verified
//
#include <hip/hip_runtime.h>

// ---------------------------------------------------------------------------
// CDNA5 (gfx1250) hetero-GraphSAGE pipeline.
//  - All matmuls: V_WMMA_F32_16X16X4_F32 (fp32 matrix path, reference numerics)
//  - Weights staged in LDS pair-interleaved so B fragments are single b64 loads
//  - Combine-kernel weights staged via TDM tensor_load_to_lds (+ s_wait_tensorcnt)
//  - Edge scatter: global_atomic_add_f32 into L2-resident (192MB) agg buffers
//  - wave32 throughout; 32-row tile per wave (2x B-fragment reuse)
// ---------------------------------------------------------------------------

typedef __attribute__((ext_vector_type(2))) float v2f;
typedef __attribute__((ext_vector_type(8))) float v8f;
typedef __attribute__((ext_vector_type(4))) unsigned int u32x4;
typedef __attribute__((ext_vector_type(4))) int i32x4;
typedef __attribute__((ext_vector_type(8))) int i32x8;

#define HDIM 64

#if defined(__has_builtin)
#if __has_builtin(__builtin_amdgcn_tensor_load_to_lds)
#define HAVE_TDM 1
#endif
#endif

__device__ __forceinline__ v8f wmma_f32(v2f a, v2f b, v8f c) {
  // 8 args: (neg_a, A, neg_b, B, c_mod, C, reuse_a, reuse_b)
  return __builtin_amdgcn_wmma_f32_16x16x4_f32(false, a, false, b, (short)0, c,
                                               false, false);
}

__device__ __forceinline__ v8f v8f_zero() {
  v8f z = {0.f, 0.f, 0.f, 0.f, 0.f, 0.f, 0.f, 0.f};
  return z;
}

#ifdef HAVE_TDM
// Issue a TDM load of a dim1 x dim0 row-major f32 tile into LDS (raw layout).
// D# per CDNA5 ISA ch.8: group0 = {count, lds_addr, global_addr[56:0], type=2},
// group1 = {data_size=4B, tensor dims, tile dims, dim0 stride}.
__device__ __forceinline__ void tdm_load_f32_tile(unsigned lds_byte_off,
                                                  const float* gsrc, int dim0,
                                                  int dim1) {
  unsigned long long ga = (unsigned long long)(uintptr_t)gsrc;
  u32x4 g0;
  g0[0] = 1u;                                  // count=1, user, no gather
  g0[1] = lds_byte_off;                        // lds_addr (bytes)
  g0[2] = (unsigned)(ga & 0xffffffffull);      // global_addr[31:0]
  g0[3] = (unsigned)((ga >> 32) & 0x1ffffffull) | (2u << 30);  // [56:32]|type=2
  i32x8 g1;
  g1[0] = 0x20000;                             // data_size=2 (4 bytes), mask=0
  g1[1] = (dim0 & 0xffff) << 16;               // tensor_dim0[15:0]
  g1[2] = ((dim0 >> 16) & 0xffff) | ((dim1 & 0xffff) << 16);  // d0 hi | d1 lo
  g1[3] = ((dim1 >> 16) & 0xffff) | ((dim0 & 0xffff) << 16);  // d1 hi | tile_d0
  g1[4] = dim1 & 0xffff;                       // tile_dim1 (tile_dim2=0)
  g1[5] = dim0;                                // tensor_dim0_stride[31:0]
  g1[6] = 0;
  g1[7] = 0;
  i32x4 z4 = {0, 0, 0, 0};
#if __clang_major__ >= 23
  i32x8 z8 = {0, 0, 0, 0, 0, 0, 0, 0};
  __builtin_amdgcn_tensor_load_to_lds(g0, g1, z4, z4, z8, 0);
#else
  __builtin_amdgcn_tensor_load_to_lds(g0, g1, z4, z4, 0);
#endif
}
#endif

// out[row, :64] = X[row, :32] @ W[32,64] + bias + emb[ids[row], :64]
// One 32-row x 64-col tile per wave; 8 waves/block -> 256 rows/block.
__global__ void __launch_bounds__(256) k_node_linear(
    const float* __restrict__ X, const float* __restrict__ W,
    const float* __restrict__ bias, const float* __restrict__ emb,
    const int* __restrict__ ids, float* __restrict__ out, int nrows) {
  __shared__ __align__(16) float sW[32 * 64];  // pair-interleaved [k/2][n][k&1]
  __shared__ float sB[64];
  for (int i = threadIdx.x; i < 32 * 64; i += 256) {
    const int k = i >> 6, n = i & 63;
    sW[(k >> 1) * 128 + n * 2 + (k & 1)] = W[i];
  }
  if (threadIdx.x < 64) sB[threadIdx.x] = bias[threadIdx.x];
  __syncthreads();

  const int wave = threadIdx.x >> 5;
  const int lane = threadIdx.x & 31;
  const int half = lane >> 4;  // 0: K {k0,k0+1}, 1: {k0+2,k0+3}
  const int ml = lane & 15;
  const int tile = (blockIdx.x * 8 + wave) * 32;
  if (tile >= nrows) return;

  int r0 = tile + ml;        if (r0 > nrows - 1) r0 = nrows - 1;
  int r1 = tile + 16 + ml;   if (r1 > nrows - 1) r1 = nrows - 1;
  const float* x0 = X + (size_t)r0 * 32;
  const float* x1 = X + (size_t)r1 * 32;

  v8f acc[2][4];
#pragma unroll
  for (int s = 0; s < 2; ++s)
#pragma unroll
    for (int t = 0; t < 4; ++t) acc[s][t] = v8f_zero();

#pragma unroll
  for (int ks = 0; ks < 8; ++ks) {  // K = 32
    const int k0 = ks * 4 + 2 * half;
    const v2f a0 = *(const v2f*)(x0 + k0);
    const v2f a1 = *(const v2f*)(x1 + k0);
#pragma unroll
    for (int t = 0; t < 4; ++t) {
      const v2f b = *(const v2f*)&sW[(k0 >> 1) * 128 + (t * 16 + ml) * 2];
      acc[0][t] = wmma_f32(a0, b, acc[0][t]);
      acc[1][t] = wmma_f32(a1, b, acc[1][t]);
    }
  }

#pragma unroll
  for (int s = 0; s < 2; ++s) {
#pragma unroll
    for (int t = 0; t < 4; ++t) {
#pragma unroll
      for (int i = 0; i < 8; ++i) {
        const int row = tile + s * 16 + i + 8 * half;  // C/D: VGPR i -> M=i|i+8
        const int col = t * 16 + ml;
        if (row < nrows) {
          const int id = ids[row];
          out[(size_t)row * HDIM + col] =
              acc[s][t][i] + sB[col] + emb[(size_t)id * HDIM + col];
        }
      }
    }
  }
}

// grid-stride zero fill
__global__ void __launch_bounds__(256) k_zero(float* __restrict__ p, size_t n) {
  size_t i = (size_t)blockIdx.x * blockDim.x + threadIdx.x;
  const size_t stride = (size_t)gridDim.x * blockDim.x;
  for (; i < n; i += stride) p[i] = 0.0f;
}

// one wave per edge: agg[sidx[e], :] += xsrc[gidx[e], :]; cnt[sidx[e]] += 1
__global__ void __launch_bounds__(256) k_scatter(
    const float* __restrict__ xsrc, const int* __restrict__ gidx,
    const int* __restrict__ sidx, float* __restrict__ agg,
    float* __restrict__ cnt, int nedges) {
  const int e = blockIdx.x * 8 + (threadIdx.x >> 5);
  const int lane = threadIdx.x & 31;
  if (e >= nedges) return;
  const int g = gidx[e];
  const int s = sidx[e];
  const float* srow = xsrc + (size_t)g * HDIM;
  float* drow = agg + (size_t)s * HDIM;
  __builtin_prefetch((const void*)(drow + lane), 1, 3);  // global_prefetch
  const float v0 = srow[lane];
  const float v1 = srow[lane + 32];
  __hip_atomic_fetch_add(drow + lane, v0, __ATOMIC_RELAXED,
                         __HIP_MEMORY_SCOPE_AGENT);
  __hip_atomic_fetch_add(drow + lane + 32, v1, __ATOMIC_RELAXED,
                         __HIP_MEMORY_SCOPE_AGENT);
  if (cnt != nullptr && lane == 0) {
    __hip_atomic_fetch_add(cnt + s, 1.0f, __ATOMIC_RELAXED,
                           __HIP_MEMORY_SCOPE_AGENT);
  }
}

// out = [mean(agg,cnt) @ wl + bl + xroot @ wr], optional ReLU. Two K=64 GEMMs
// fused into the same WMMA accumulators; weights staged via TDM when present.
__global__ void __launch_bounds__(256) k_sage_combine(
    const float* __restrict__ agg, const float* __restrict__ cnt,
    const float* __restrict__ xroot, const float* __restrict__ wl,
    const float* __restrict__ bl, const float* __restrict__ wr,
    float* __restrict__ out, int nrows, int doRelu) {
  __shared__ __align__(16) float sRawL[64 * 64];
  __shared__ __align__(16) float sRawR[64 * 64];
  __shared__ __align__(16) float sWl[64 * 64];  // pair-interleaved
  __shared__ __align__(16) float sWr[64 * 64];  // pair-interleaved
  __shared__ float sB[64];

#ifdef HAVE_TDM
  if (threadIdx.x < 32) {  // one wave issues both tile DMAs
    tdm_load_f32_tile((unsigned)(uintptr_t)(void*)sRawL, wl, 64, 64);
    tdm_load_f32_tile((unsigned)(uintptr_t)(void*)sRawR, wr, 64, 64);
    __builtin_amdgcn_s_wait_tensorcnt(0);
  }
#else
  for (int i = threadIdx.x; i < 64 * 64; i += 256) {
    sRawL[i] = wl[i];
    sRawR[i] = wr[i];
  }
#endif
  if (threadIdx.x < 64) sB[threadIdx.x] = bl[threadIdx.x];
  __syncthreads();
  // repack raw row-major -> pair-interleaved [k/2][n][k&1]
  for (int i = threadIdx.x; i < 64 * 64; i += 256) {
    const int k = i >> 6, n = i & 63;
    const int pi = (k >> 1) * 128 + n * 2 + (k & 1);
    sWl[pi] = sRawL[i];
    sWr[pi] = sRawR[i];
  }
  __syncthreads();

  const int wave = threadIdx.x >> 5;
  const int lane = threadIdx.x & 31;
  const int half = lane >> 4;
  const int ml = lane & 15;
  const int tile = (blockIdx.x * 8 + wave) * 32;
  if (tile >= nrows) return;

  int r0 = tile + ml;        if (r0 > nrows - 1) r0 = nrows - 1;
  int r1 = tile + 16 + ml;   if (r1 > nrows - 1) r1 = nrows - 1;
  const float rc0 = 1.0f / fmaxf(cnt[r0], 1.0f);
  const float rc1 = 1.0f / fmaxf(cnt[r1], 1.0f);
  const float* a0 = agg + (size_t)r0 * HDIM;
  const float* a1 = agg + (size_t)r1 * HDIM;
  const float* x0 = xroot + (size_t)r0 * HDIM;
  const float* x1 = xroot + (size_t)r1 * HDIM;

  v8f acc[2][4];
#pragma unroll
  for (int s = 0; s < 2; ++s)
#pragma unroll
    for (int t = 0; t < 4; ++t) acc[s][t] = v8f_zero();

#pragma unroll
  for (int ks = 0; ks < 16; ++ks) {  // K = 64
    const int k0 = ks * 4 + 2 * half;
    const v2f am0 = (*(const v2f*)(a0 + k0)) * rc0;
    const v2f am1 = (*(const v2f*)(a1 + k0)) * rc1;
    const v2f ar0 = *(const v2f*)(x0 + k0);
    const v2f ar1 = *(const v2f*)(x1 + k0);
#pragma unroll
    for (int t = 0; t < 4; ++t) {
      const int bi = (k0 >> 1) * 128 + (t * 16 + ml) * 2;
      const v2f b0 = *(const v2f*)&sWl[bi];
      acc[0][t] = wmma_f32(am0, b0, acc[0][t]);
      acc[1][t] = wmma_f32(am1, b0, acc[1][t]);
      const v2f b1 = *(const v2f*)&sWr[bi];
      acc[0][t] = wmma_f32(ar0, b1, acc[0][t]);
      acc[1][t] = wmma_f32(ar1, b1, acc[1][t]);
    }
  }

#pragma unroll
  for (int s = 0; s < 2; ++s) {
#pragma unroll
    for (int t = 0; t < 4; ++t) {
#pragma unroll
      for (int i = 0; i < 8; ++i) {
        const int row = tile + s * 16 + i + 8 * half;
        const int col = t * 16 + ml;
        if (row < nrows) {
          float v = acc[s][t][i] + sB[col];
          if (doRelu) v = fmaxf(v, 0.0f);
          out[(size_t)row * HDIM + col] = v;
        }
      }
    }
  }
}

// one wave per label edge: out[e] = dot(hu[ls[e],:64], hm[ld[e],:64])
__global__ void __launch_bounds__(256) k_dot(
    const float* __restrict__ hu, const float* __restrict__ hm,
    const int* __restrict__ ls, const int* __restrict__ ld,
    float* __restrict__ out, int ne) {
  const int e = blockIdx.x * 8 + (threadIdx.x >> 5);
  const int lane = threadIdx.x & 31;
  if (e >= ne) return;
  const float* u = hu + (size_t)ls[e] * HDIM;
  const float* m = hm + (size_t)ld[e] * HDIM;
  float v = u[lane] * m[lane] + u[lane + 32] * m[lane + 32];
#pragma unroll
  for (int off = 16; off > 0; off >>= 1) v += __shfl_xor(v, off, 32);
  if (lane == 0) out[e] = v;
}

extern "C" void kernel_launch(void* const* d_in, const int* in_sizes, int n_in,
                              void* d_out, int out_size, void* d_ws,
                              size_t ws_size, hipStream_t stream) {
  const float* x_user = (const float*)d_in[0];
  const float* x_movie = (const float*)d_in[1];
  const float* user_lin_w = (const float*)d_in[2];
  const float* user_lin_b = (const float*)d_in[3];
  const float* movie_lin_w = (const float*)d_in[4];
  const float* movie_lin_b = (const float*)d_in[5];
  const float* user_emb = (const float*)d_in[6];
  const float* movie_emb = (const float*)d_in[7];
  const float* c1_um_wl = (const float*)d_in[8];
  const float* c1_um_bl = (const float*)d_in[9];
  const float* c1_um_wr = (const float*)d_in[10];
  const float* c1_mu_wl = (const float*)d_in[11];
  const float* c1_mu_bl = (const float*)d_in[12];
  const float* c1_mu_wr = (const float*)d_in[13];
  const float* c2_um_wl = (const float*)d_in[14];
  const float* c2_um_bl = (const float*)d_in[15];
  const float* c2_um_wr = (const float*)d_in[16];
  const float* c2_mu_wl = (const float*)d_in[17];
  const float* c2_mu_bl = (const float*)d_in[18];
  const float* c2_mu_wr = (const float*)d_in[19];
  const int* user_ids = (const int*)d_in[20];
  const int* movie_ids = (const int*)d_in[21];
  const int* edge_src = (const int*)d_in[22];
  const int* edge_dst = (const int*)d_in[23];
  const int* label_src = (const int*)d_in[24];
  const int* label_dst = (const int*)d_in[25];

  const int NU = in_sizes[20];
  const int NM = in_sizes[21];
  const int E = in_sizes[22];
  const int EL = in_sizes[24];

  // workspace layout (floats); agg/cnt contiguous so zero passes are single
  float* ws = (float*)d_ws;
  size_t o = 0;
  float* xu = ws + o;   o += (size_t)NU * HDIM;
  float* xm = ws + o;   o += (size_t)NM * HDIM;
  float* aggU = ws + o; o += (size_t)NU * HDIM;
  float* aggM = ws + o; o += (size_t)NM * HDIM;
  float* cntU = ws + o; o += (size_t)NU;
  float* cntM = ws + o; o += (size_t)NM;
  float* h1u = ws + o;  o += (size_t)NU * HDIM;
  float* h1m = ws + o;  o += (size_t)NM * HDIM;
  const size_t aggCntSpan = (size_t)(NU + NM) * HDIM + NU + NM;
  const size_t aggSpan = (size_t)(NU + NM) * HDIM;

  const int blkU = (NU + 255) / 256;  // 8 waves/block x 32 rows/wave
  const int blkM = (NM + 255) / 256;
  const int blkE = (E + 7) / 8;       // wave per edge
  const int blkL = (EL + 7) / 8;

  // 1) node input linears (+bias +emb gather)
  k_node_linear<<<blkU, 256, 0, stream>>>(x_user, user_lin_w, user_lin_b,
                                          user_emb, user_ids, xu, NU);
  k_node_linear<<<blkM, 256, 0, stream>>>(x_movie, movie_lin_w, movie_lin_b,
                                          movie_emb, movie_ids, xm, NM);

  // 2) layer 1 scatter-mean (counts computed once, reused in layer 2)
  k_zero<<<2048, 256, 0, stream>>>(aggU, aggCntSpan);
  k_scatter<<<blkE, 256, 0, stream>>>(xm, edge_dst, edge_src, aggU, cntU, E);
  k_scatter<<<blkE, 256, 0, stream>>>(xu, edge_src, edge_dst, aggM, cntM, E);
  k_sage_combine<<<blkU, 256, 0, stream>>>(aggU, cntU, xu, c1_mu_wl, c1_mu_bl,
                                           c1_mu_wr, h1u, NU, 1);
  k_sage_combine<<<blkM, 256, 0, stream>>>(aggM, cntM, xm, c1_um_wl, c1_um_bl,
                                           c1_um_wr, h1m, NM, 1);

  // 3) layer 2 (counts already known; zero agg only)
  k_zero<<<2048, 256, 0, stream>>>(aggU, aggSpan);
  k_scatter<<<blkE, 256, 0, stream>>>(h1m, edge_dst, edge_src, aggU, nullptr, E);
  k_scatter<<<blkE, 256, 0, stream>>>(h1u, edge_src, edge_dst, aggM, nullptr, E);
  k_sage_combine<<<blkU, 256, 0, stream>>>(aggU, cntU, h1u, c2_mu_wl, c2_mu_bl,
                                           c2_mu_wr, xu, NU, 0);  // h2u -> xu
  k_sage_combine<<<blkM, 256, 0, stream>>>(aggM, cntM, h1m, c2_um_wl, c2_um_bl,
                                           c2_um_wr, xm, NM, 0);  // h2m -> xm

  // 4) edge dot-product classifier
  k_dot<<<blkL, 256, 0, stream>>>(xu, xm, label_src, label_dst, (float*)d_out,
                                  EL);
}